// DecodeAttention_82085414961546
// MI455X (gfx1250) — compile-verified
//
#include <hip/hip_runtime.h>
#include <hip/hip_bf16.h>

typedef float v2f __attribute__((ext_vector_type(2)));
typedef float v8f __attribute__((ext_vector_type(8)));

#define BATCH 32
#define SEQ   4095
#define EMB   1024
#define HEADS 16
#define HD    64
#define SLEN  4096   // SEQ + 1

// ---------------------------------------------------------------------------
// GEMM: out[32 x N] = A[32 x K] * W[K x N] + bias[N], f32 WMMA 16x16x4.
// One wave (32 threads) per 16-wide N tile; two M-tiles (rows 0-15, 16-31).
// ---------------------------------------------------------------------------
__global__ __launch_bounds__(32)
void gemm32_wmma_f32(const float* __restrict__ A,
                     const float* __restrict__ W,
                     const float* __restrict__ bias,
                     float* __restrict__ out,
                     int N, int K) {
    const int n0   = blockIdx.x * 16;
    const int lane = threadIdx.x;      // 0..31, full wave, no divergence
    const int half = lane >> 4;        // 0: K lanes {0,1}; 1: K lanes {2,3}
    const int l16  = lane & 15;

    v8f acc0 = {};   // rows 0..15
    v8f acc1 = {};   // rows 16..31

    for (int k0 = 0; k0 < K; k0 += 4) {
        const int ka = k0 + 2 * half;
        // B tile (4x16): VGPR0 = row ka, VGPR1 = row ka+1, col n0+l16
        v2f b;
        b.x = W[(size_t)(ka + 0) * N + n0 + l16];
        b.y = W[(size_t)(ka + 1) * N + n0 + l16];
        // A tiles (16x4): lane l16 = matrix row
        v2f a0, a1;
        a0.x = A[(size_t)(l16)      * K + ka + 0];
        a0.y = A[(size_t)(l16)      * K + ka + 1];
        a1.x = A[(size_t)(l16 + 16) * K + ka + 0];
        a1.y = A[(size_t)(l16 + 16) * K + ka + 1];

        acc0 = __builtin_amdgcn_wmma_f32_16x16x4_f32(false, a0, false, b,
                                                     (short)0, acc0, false, false);
        acc1 = __builtin_amdgcn_wmma_f32_16x16x4_f32(false, a1, false, b,
                                                     (short)0, acc1, false, false);
    }

    const int   col = n0 + l16;
    const float bv  = bias[col];
#pragma unroll
    for (int j = 0; j < 8; ++j) {
        const int row = j + 8 * half;              // C/D layout: M = j + 8*(lane>=16)
        out[(size_t)(row)      * N + col] = acc0[j] + bv;
        out[(size_t)(row + 16) * N + col] = acc1[j] + bv;
    }
}

// ---------------------------------------------------------------------------
// last[b] = largest s with mask[b,s] == rowmax(mask[b,:])
// ---------------------------------------------------------------------------
__global__ __launch_bounds__(256)
void find_last_kernel(const int* __restrict__ mask, int* __restrict__ last) {
    const int b   = blockIdx.x;
    const int tid = threadIdx.x;
    __shared__ int red[256];

    int mmax = -2147483647;
    for (int s = tid; s < SLEN; s += 256)
        mmax = max(mmax, mask[(size_t)b * SLEN + s]);
    red[tid] = mmax;
    __syncthreads();
    for (int off = 128; off > 0; off >>= 1) {
        if (tid < off) red[tid] = max(red[tid], red[tid + off]);
        __syncthreads();
    }
    const int M = red[0];
    __syncthreads();

    int li = -1;
    for (int s = tid; s < SLEN; s += 256)
        if (mask[(size_t)b * SLEN + s] == M) li = max(li, s);
    red[tid] = li;
    __syncthreads();
    for (int off = 128; off > 0; off >>= 1) {
        if (tid < off) red[tid] = max(red[tid], red[tid + off]);
        __syncthreads();
    }
    if (tid == 0) last[b] = red[0];
}

// ---------------------------------------------------------------------------
// Cache rebuild (+ optional fused score computation while K row is live in
// registers). One block per (s,b) row: 256 thr x float4 = 1024 floats each
// for K and V. When do_scores!=0 also emits scores[b,h,s] (16 heads).
// ---------------------------------------------------------------------------
__global__ __launch_bounds__(256)
void build_cache_kernel(const float* __restrict__ qkv,      // 32 x 3072
                        const float* __restrict__ key_cache,
                        const float* __restrict__ value_cache,
                        const int*   __restrict__ last,
                        const int*   __restrict__ mask,
                        const float* __restrict__ attn_bias, // row 0 used
                        float* __restrict__ new_k,
                        float* __restrict__ new_v,
                        float* __restrict__ scores,          // 32 x 16 x 4096
                        int do_scores) {
    const int s = blockIdx.x;   // 0..4095
    const int b = blockIdx.y;   // 0..31
    const int t = threadIdx.x;  // 0..255 -> float4 index
    const int lb = last[b];

    const size_t dst = ((size_t)b * SLEN + s) * EMB;
    float4 kv, vv;
    if (s == lb) {
        kv = ((const float4*)(qkv + (size_t)b * 3 * EMB + EMB))[t];      // k slice
        vv = ((const float4*)(qkv + (size_t)b * 3 * EMB + 2 * EMB))[t];  // v slice
    } else if (s < SEQ) {
        const size_t src = ((size_t)b * SEQ + s) * EMB;
        kv = ((const float4*)(key_cache   + src))[t];
        vv = ((const float4*)(value_cache + src))[t];
    } else {
        kv = make_float4(0.f, 0.f, 0.f, 0.f);
        vv = kv;
    }
    ((float4*)(new_k + dst))[t] = kv;
    ((float4*)(new_v + dst))[t] = vv;

    if (do_scores) {
        // q . k partial: thread t covers elements 4t..4t+3 (head = t>>4)
        __shared__ float part[256];
        const float4 q4 = ((const float4*)(qkv + (size_t)b * 3 * EMB))[t];
        part[t] = q4.x * kv.x + q4.y * kv.y + q4.z * kv.z + q4.w * kv.w;
        __syncthreads();
        if (t < HEADS) {
            float acc = 0.f;
#pragma unroll
            for (int i = 0; i < 16; ++i) acc += part[t * 16 + i];
            float sc = acc * 0.125f + attn_bias[s];     // 1/sqrt(64)
            if (mask[(size_t)b * SLEN + s] == 0) sc = -1.0e9f;
            scores[((size_t)b * HEADS + t) * SLEN + s] = sc;
        }
    }
}

// ---------------------------------------------------------------------------
// Softmax + weighted-V from precomputed scores (fused path).
// ---------------------------------------------------------------------------
__global__ __launch_bounds__(256)
void attn_from_scores_kernel(const float* __restrict__ scores,
                             const float* __restrict__ new_v,
                             float* __restrict__ ctx) {     // 32 x 1024
    const int h   = blockIdx.x;
    const int b   = blockIdx.y;
    const int tid = threadIdx.x;

    __shared__ float ssc[SLEN];
    __shared__ float red[256];
    __shared__ float part[4][HD];

    const float* srow = scores + ((size_t)b * HEADS + h) * SLEN;
    float lm = -3.4e38f;
    for (int s = tid; s < SLEN; s += 256) {
        float v = srow[s];
        ssc[s] = v;
        lm = fmaxf(lm, v);
    }
    red[tid] = lm;
    __syncthreads();
    for (int off = 128; off > 0; off >>= 1) {
        if (tid < off) red[tid] = fmaxf(red[tid], red[tid + off]);
        __syncthreads();
    }
    const float m = red[0];
    __syncthreads();

    float ls = 0.f;
    for (int s = tid; s < SLEN; s += 256) {
        float e = __expf(ssc[s] - m);
        ssc[s] = e;
        ls += e;
    }
    red[tid] = ls;
    __syncthreads();
    for (int off = 128; off > 0; off >>= 1) {
        if (tid < off) red[tid] += red[tid + off];
        __syncthreads();
    }
    const float inv = 1.0f / red[0];
    __syncthreads();

    const int d     = tid & (HD - 1);
    const int chunk = tid >> 6;
    float acc = 0.f;
    for (int s = chunk; s < SLEN; s += 4)
        acc += ssc[s] * new_v[((size_t)b * SLEN + s) * EMB + h * HD + d];
    part[chunk][d] = acc;
    __syncthreads();
    if (chunk == 0)
        ctx[(size_t)b * EMB + h * HD + d] =
            (part[0][d] + part[1][d] + part[2][d] + part[3][d]) * inv;
}

// ---------------------------------------------------------------------------
// Full attention fallback (recomputes scores from new_k) — used only when the
// workspace is too small for the scores buffer.
// ---------------------------------------------------------------------------
__global__ __launch_bounds__(256)
void attn_kernel(const float* __restrict__ qkv,
                 const float* __restrict__ new_k,
                 const float* __restrict__ new_v,
                 const int*   __restrict__ mask,
                 const float* __restrict__ attn_bias,
                 float* __restrict__ ctx) {
    const int h   = blockIdx.x;
    const int b   = blockIdx.y;
    const int tid = threadIdx.x;

    __shared__ float qsh[HD];
    __shared__ float ssc[SLEN];
    __shared__ float red[256];
    __shared__ float part[4][HD];

    if (tid < HD) qsh[tid] = qkv[(size_t)b * 3 * EMB + h * HD + tid];
    __syncthreads();

    for (int s = tid; s < SLEN; s += 256) {
        const float4* kp = (const float4*)(new_k + ((size_t)b * SLEN + s) * EMB + h * HD);
        float acc = 0.f;
#pragma unroll
        for (int i = 0; i < HD / 4; ++i) {
            float4 kk = kp[i];
            acc += qsh[4 * i + 0] * kk.x + qsh[4 * i + 1] * kk.y +
                   qsh[4 * i + 2] * kk.z + qsh[4 * i + 3] * kk.w;
        }
        float sc = acc * 0.125f + attn_bias[s];
        if (mask[(size_t)b * SLEN + s] == 0) sc = -1.0e9f;
        ssc[s] = sc;
    }
    __syncthreads();

    float lm = -3.4e38f;
    for (int s = tid; s < SLEN; s += 256) lm = fmaxf(lm, ssc[s]);
    red[tid] = lm;
    __syncthreads();
    for (int off = 128; off > 0; off >>= 1) {
        if (tid < off) red[tid] = fmaxf(red[tid], red[tid + off]);
        __syncthreads();
    }
    const float m = red[0];
    __syncthreads();

    float ls = 0.f;
    for (int s = tid; s < SLEN; s += 256) {
        float e = __expf(ssc[s] - m);
        ssc[s] = e;
        ls += e;
    }
    red[tid] = ls;
    __syncthreads();
    for (int off = 128; off > 0; off >>= 1) {
        if (tid < off) red[tid] += red[tid + off];
        __syncthreads();
    }
    const float inv = 1.0f / red[0];
    __syncthreads();

    const int d     = tid & (HD - 1);
    const int chunk = tid >> 6;
    float acc = 0.f;
    for (int s = chunk; s < SLEN; s += 4)
        acc += ssc[s] * new_v[((size_t)b * SLEN + s) * EMB + h * HD + d];
    part[chunk][d] = acc;
    __syncthreads();
    if (chunk == 0)
        ctx[(size_t)b * EMB + h * HD + d] =
            (part[0][d] + part[1][d] + part[2][d] + part[3][d]) * inv;
}

// ---------------------------------------------------------------------------
extern "C" void kernel_launch(void* const* d_in, const int* in_sizes, int n_in,
                              void* d_out, int out_size, void* d_ws, size_t ws_size,
                              hipStream_t stream) {
    const float* hidden    = (const float*)d_in[0];  // 32 x 1 x 1024
    const float* key_cache = (const float*)d_in[1];  // 32 x 4095 x 1024
    const float* val_cache = (const float*)d_in[2];
    const int*   mask      = (const int*)  d_in[3];  // 32 x 4096
    const float* c_attn_w  = (const float*)d_in[4];  // 1024 x 3072
    const float* c_attn_b  = (const float*)d_in[5];  // 3072
    const float* c_proj_w  = (const float*)d_in[6];  // 1024 x 1024
    const float* c_proj_b  = (const float*)d_in[7];  // 1024
    const float* attn_bias = (const float*)d_in[8];  // 4096 x 4096 (row 0 used)

    float* out   = (float*)d_out;                    // 32 x 1024
    float* new_k = out + (size_t)BATCH * EMB;        // 32 x 4096 x 1024
    float* new_v = new_k + (size_t)BATCH * SLEN * EMB;

    // Workspace layout
    const size_t qkvBytes    = (size_t)BATCH * 3 * EMB * sizeof(float); // 393216
    const size_t lastOff     = qkvBytes;                                // 128 B
    const size_t ctxOff      = qkvBytes + 256;
    const size_t ctxBytes    = (size_t)BATCH * EMB * sizeof(float);     // 131072
    const size_t scoresOff   = ctxOff + ctxBytes;
    const size_t scoresBytes = (size_t)BATCH * HEADS * SLEN * sizeof(float); // 8 MB

    char*  ws     = (char*)d_ws;
    float* qkv    = (float*)ws;
    int*   last   = (int*)(ws + lastOff);
    float* ctx    = (float*)(ws + ctxOff);
    float* scores = (float*)(ws + scoresOff);
    const int fused = (ws_size >= scoresOff + scoresBytes) ? 1 : 0;

    // 1) QKV projection (WMMA f32)
    gemm32_wmma_f32<<<3 * EMB / 16, 32, 0, stream>>>(hidden, c_attn_w, c_attn_b,
                                                     qkv, 3 * EMB, EMB);
    // 2) last-valid index per batch
    find_last_kernel<<<BATCH, 256, 0, stream>>>(mask, last);
    // 3) rebuild caches; fused path also emits scores while K rows are live
    build_cache_kernel<<<dim3(SLEN, BATCH), 256, 0, stream>>>(
        qkv, key_cache, val_cache, last, mask, attn_bias,
        new_k, new_v, fused ? scores : (float*)ws, fused);
    // 4) softmax + weighted V
    if (fused) {
        attn_from_scores_kernel<<<dim3(HEADS, BATCH), 256, 0, stream>>>(scores, new_v, ctx);
    } else {
        attn_kernel<<<dim3(HEADS, BATCH), 256, 0, stream>>>(qkv, new_k, new_v, mask,
                                                            attn_bias, ctx);
    }
    // 5) output projection (WMMA f32)
    gemm32_wmma_f32<<<EMB / 16, 32, 0, stream>>>(ctx, c_proj_w, c_proj_b,
                                                 out, EMB, EMB);
}